// ResidualGCN_67551245631652
// MI455X (gfx1250) — compile-verified
//
#include <hip/hip_runtime.h>
#include <hip/hip_bf16.h>

// ---------------------------------------------------------------------------
// Residual GCN for MI455X (gfx1250, wave32).
//   deg/dis -> [WMMA f32 GEMM -> scatter-add (atomics) -> finish]*3 -> edge MLP
// Node GEMMs use V_WMMA_F32_16X16X4_F32 (fp32-preserving CDNA5 matrix path).
// ---------------------------------------------------------------------------

typedef float v2f __attribute__((ext_vector_type(2)));
typedef float v8f __attribute__((ext_vector_type(8)));

#define NNODES 100000
#define NEDGES 3200000

// ---------------------------------------------------------------------------
// Degree / normalization
// ---------------------------------------------------------------------------
__global__ void deg_count_kernel(const int* __restrict__ dst, float* __restrict__ deg, int E) {
    int e = blockIdx.x * blockDim.x + threadIdx.x;
    if (e < E) atomicAdd(&deg[dst[e]], 1.0f);
}

__global__ void make_dis_kernel(float* __restrict__ deg, int n) {
    int i = blockIdx.x * blockDim.x + threadIdx.x;
    if (i < n) deg[i] = rsqrtf(deg[i] + 1.0f);   // self-loop included
}

// ---------------------------------------------------------------------------
// WMMA fp32 GEMM: Y[nrows x F] = X[nrows x K] @ W[K x F]   (row-major)
// One wave32 computes one 16x16 output tile, K stepped by 4 per v_wmma.
// A frag (16x4 f32, 2 VGPR): lanes 0-15 -> (M=lane, K=k+0/k+1),
//                            lanes 16-31 -> (M=lane-16, K=k+2/k+3).
// B frag mirrors A's K split; C/D: VGPR r holds (M=r | M=r+8) x (N=lane&15).
// ---------------------------------------------------------------------------
template <int K, int F>
__global__ void wmma_gemm_kernel(const float* __restrict__ X,
                                 const float* __restrict__ W,
                                 float* __restrict__ Y, int nrows) {
    const int wave = (blockIdx.x * blockDim.x + threadIdx.x) >> 5;
    const int lane = threadIdx.x & 31;
    const int tilesN = F / 16;
    const int totalTiles = (nrows / 16) * tilesN;
    if (wave >= totalTiles) return;            // wave-uniform: EXEC stays all-1s

    const int tileM = wave / tilesN;
    const int tileN = wave % tilesN;
    const int row0 = tileM * 16;
    const int col0 = tileN * 16;
    const int half = lane >> 4;                // 0: K lo pair, 1: K hi pair
    const int l15  = lane & 15;

    const float* __restrict__ Xrow = X + (size_t)(row0 + l15) * K + half * 2;
    const float* __restrict__ Wcol = W + col0 + l15;

    v8f acc = {};
#pragma unroll 4
    for (int k = 0; k < K; k += 4) {
        v2f a, b;
        a.x = Xrow[k];
        a.y = Xrow[k + 1];
        b.x = Wcol[(size_t)(k + half * 2) * F];
        b.y = Wcol[(size_t)(k + half * 2 + 1) * F];
        acc = __builtin_amdgcn_wmma_f32_16x16x4_f32(
            /*neg_a=*/false, a, /*neg_b=*/false, b,
            /*c_mod=*/(short)0, acc, /*reuse_a=*/false, /*reuse_b=*/false);
    }

    float* __restrict__ Yt = Y + col0 + l15;
    const int rbase = row0 + half * 8;
#pragma unroll
    for (int r = 0; r < 8; ++r)
        Yt[(size_t)(rbase + r) * F] = acc[r];
}

// ---------------------------------------------------------------------------
// Edge scatter: agg[dst] += hlin[src] * dis[src]*dis[dst]   (atomic fadd)
// ---------------------------------------------------------------------------
template <int F>
__global__ void scatter_kernel(const float* __restrict__ hlin,
                               const int* __restrict__ src,
                               const int* __restrict__ dst,
                               const float* __restrict__ dis,
                               float* __restrict__ agg, int E) {
    int e = blockIdx.x * blockDim.x + threadIdx.x;
    if (e >= E) return;
    const int s = src[e], d = dst[e];
    const float nrm = dis[s] * dis[d];
    const float4* __restrict__ hs = (const float4*)(hlin + (size_t)s * F);
    float* __restrict__ ad = agg + (size_t)d * F;
#pragma unroll
    for (int i = 0; i < F / 4; ++i) {
        float4 v = hs[i];
        atomicAdd(ad + 4 * i + 0, v.x * nrm);
        atomicAdd(ad + 4 * i + 1, v.y * nrm);
        atomicAdd(ad + 4 * i + 2, v.z * nrm);
        atomicAdd(ad + 4 * i + 3, v.w * nrm);
    }
}

// ---------------------------------------------------------------------------
// Per-node finish: out = [relu](agg + hlin*dis^2 + b) [+ resid]
// ---------------------------------------------------------------------------
template <int F, bool RELU, bool RESID>
__global__ void finish_kernel(const float* __restrict__ agg,
                              const float* __restrict__ hlin,
                              const float* __restrict__ dis,
                              const float* __restrict__ bias,
                              const float* __restrict__ resid,
                              float* __restrict__ out, int nrows) {
    int i = blockIdx.x * blockDim.x + threadIdx.x;
    if (i >= nrows * F) return;
    const int f = i % F;
    const int n = i / F;
    float d = dis[n];
    float v = agg[i] + hlin[i] * (d * d) + bias[f];
    if (RELU) v = fmaxf(v, 0.0f);
    if (RESID) v += resid[i];
    out[i] = v;
}

// ---------------------------------------------------------------------------
// Edge MLP: ef = [h3[src], h3[dst]] (32) -> relu(fc1) (16) -> fc2 (16)
//           -> log_softmax. Weights staged in LDS. One thread per edge.
// ---------------------------------------------------------------------------
__global__ void edge_mlp_kernel(const float* __restrict__ h3,
                                const int* __restrict__ src,
                                const int* __restrict__ dst,
                                const float* __restrict__ fc1W,
                                const float* __restrict__ fc1b,
                                const float* __restrict__ fc2W,
                                const float* __restrict__ fc2b,
                                float* __restrict__ out, int E) {
    __shared__ float sW1[32 * 16];
    __shared__ float sW2[16 * 16];
    __shared__ float sb1[16];
    __shared__ float sb2[16];
    for (int i = threadIdx.x; i < 32 * 16; i += blockDim.x) sW1[i] = fc1W[i];
    for (int i = threadIdx.x; i < 16 * 16; i += blockDim.x) sW2[i] = fc2W[i];
    if (threadIdx.x < 16) {
        sb1[threadIdx.x] = fc1b[threadIdx.x];
        sb2[threadIdx.x] = fc2b[threadIdx.x];
    }
    __syncthreads();

    const int e = blockIdx.x * blockDim.x + threadIdx.x;
    if (e >= E) return;

    const float4* __restrict__ ps = (const float4*)(h3 + (size_t)src[e] * 16);
    const float4* __restrict__ pd = (const float4*)(h3 + (size_t)dst[e] * 16);
    __builtin_prefetch(pd, 0, 0);              // global_prefetch_b8 on dst row

    float ef[32];
#pragma unroll
    for (int i = 0; i < 4; ++i) {
        float4 v = ps[i];
        ef[4 * i + 0] = v.x; ef[4 * i + 1] = v.y;
        ef[4 * i + 2] = v.z; ef[4 * i + 3] = v.w;
    }
#pragma unroll
    for (int i = 0; i < 4; ++i) {
        float4 v = pd[i];
        ef[16 + 4 * i + 0] = v.x; ef[16 + 4 * i + 1] = v.y;
        ef[16 + 4 * i + 2] = v.z; ef[16 + 4 * i + 3] = v.w;
    }

    float y1[16];
#pragma unroll
    for (int j = 0; j < 16; ++j) y1[j] = sb1[j];
#pragma unroll 4
    for (int k = 0; k < 32; ++k) {
        const float a = ef[k];
        const float* wrow = &sW1[k * 16];
#pragma unroll
        for (int j = 0; j < 16; ++j) y1[j] = fmaf(a, wrow[j], y1[j]);
    }
#pragma unroll
    for (int j = 0; j < 16; ++j) y1[j] = fmaxf(y1[j], 0.0f);

    float y2[16];
#pragma unroll
    for (int j = 0; j < 16; ++j) y2[j] = sb2[j];
#pragma unroll 4
    for (int k = 0; k < 16; ++k) {
        const float a = y1[k];
        const float* wrow = &sW2[k * 16];
#pragma unroll
        for (int j = 0; j < 16; ++j) y2[j] = fmaf(a, wrow[j], y2[j]);
    }

    // log_softmax over the 16 classes
    float m = y2[0];
#pragma unroll
    for (int j = 1; j < 16; ++j) m = fmaxf(m, y2[j]);
    float s = 0.0f;
#pragma unroll
    for (int j = 0; j < 16; ++j) s += __expf(y2[j] - m);
    const float lse = m + __logf(s);

    float* __restrict__ po = out + (size_t)e * 16;
#pragma unroll
    for (int j = 0; j < 16; ++j) po[j] = y2[j] - lse;
}

// ---------------------------------------------------------------------------
// Host-side orchestration (graph-capture safe: only async ops on `stream`)
// ---------------------------------------------------------------------------
extern "C" void kernel_launch(void* const* d_in, const int* in_sizes, int n_in,
                              void* d_out, int out_size, void* d_ws, size_t ws_size,
                              hipStream_t stream) {
    const float* x    = (const float*)d_in[0];   // [N,128]
    const int*   ei   = (const int*)d_in[1];     // [2,E]
    const float* W1   = (const float*)d_in[2];   // [128,64]
    const float* b1   = (const float*)d_in[3];
    const float* W2   = (const float*)d_in[4];   // [64,64]
    const float* b2   = (const float*)d_in[5];
    const float* W3   = (const float*)d_in[6];   // [64,16]
    const float* b3   = (const float*)d_in[7];
    const float* fc1W = (const float*)d_in[8];   // [32,16]
    const float* fc1b = (const float*)d_in[9];
    const float* fc2W = (const float*)d_in[10];  // [16,16]
    const float* fc2b = (const float*)d_in[11];

    const int N = in_sizes[0] / 128;             // 100000 (multiple of 16)
    const int E = in_sizes[1] / 2;               // 3200000
    const int* src = ei;
    const int* dst = ei + E;

    // Workspace layout (floats)
    float* ws   = (float*)d_ws;
    float* dis  = ws;                            // N
    float* h1   = dis  + N;                      // N*64  conv1 out (residual)
    float* hlin = h1   + (size_t)N * 64;         // N*64  linear-part scratch
    float* agg  = hlin + (size_t)N * 64;         // N*64  scatter accumulator
    float* hcur = agg  + (size_t)N * 64;         // N*64  conv2 out
    float* h3   = hcur + (size_t)N * 64;         // N*16  conv3 out
    (void)ws_size; (void)n_in; (void)out_size;

    const int TB = 256;
    const int ebk = (E + TB - 1) / TB;

    // --- degree / dis ---
    hipMemsetAsync(dis, 0, (size_t)N * sizeof(float), stream);
    deg_count_kernel<<<ebk, TB, 0, stream>>>(dst, dis, E);
    make_dis_kernel<<<(N + TB - 1) / TB, TB, 0, stream>>>(dis, N);

    // --- layer 1: hlin = x@W1 ; aggregate ; relu ---
    {
        const int waves = (N / 16) * (64 / 16);
        wmma_gemm_kernel<128, 64><<<(waves + 7) / 8, TB, 0, stream>>>(x, W1, hlin, N);
        hipMemsetAsync(agg, 0, (size_t)N * 64 * sizeof(float), stream);
        scatter_kernel<64><<<ebk, TB, 0, stream>>>(hlin, src, dst, dis, agg, E);
        finish_kernel<64, true, false><<<((N * 64) + TB - 1) / TB, TB, 0, stream>>>(
            agg, hlin, dis, b1, nullptr, h1, N);
    }

    // --- layer 2: hlin = h1@W2 ; aggregate ; relu + residual(h1) ---
    {
        const int waves = (N / 16) * (64 / 16);
        wmma_gemm_kernel<64, 64><<<(waves + 7) / 8, TB, 0, stream>>>(h1, W2, hlin, N);
        hipMemsetAsync(agg, 0, (size_t)N * 64 * sizeof(float), stream);
        scatter_kernel<64><<<ebk, TB, 0, stream>>>(hlin, src, dst, dis, agg, E);
        finish_kernel<64, true, true><<<((N * 64) + TB - 1) / TB, TB, 0, stream>>>(
            agg, hlin, dis, b2, h1, hcur, N);
    }

    // --- layer 3: hlin(N*16) = hcur@W3 ; aggregate ; no relu ---
    {
        const int waves = (N / 16) * (16 / 16);
        wmma_gemm_kernel<64, 16><<<(waves + 7) / 8, TB, 0, stream>>>(hcur, W3, hlin, N);
        hipMemsetAsync(agg, 0, (size_t)N * 16 * sizeof(float), stream);
        scatter_kernel<16><<<ebk, TB, 0, stream>>>(hlin, src, dst, dis, agg, E);
        finish_kernel<16, false, false><<<((N * 16) + TB - 1) / TB, TB, 0, stream>>>(
            agg, hlin, dis, b3, nullptr, h3, N);
    }

    // --- edge MLP + log_softmax ---
    edge_mlp_kernel<<<ebk, TB, 0, stream>>>(h3, src, dst, fc1W, fc1b, fc2W, fc2b,
                                            (float*)d_out, E);
}